// TTN_Pool_2by2to1_40638980555112
// MI455X (gfx1250) — compile-verified
//
#include <hip/hip_runtime.h>

// CDNA5 / gfx1250 WMMA fragment types (wave32)
typedef __attribute__((ext_vector_type(2))) float v2f;  // A/B frag of 16x16x4 f32 (2 VGPRs)
typedef __attribute__((ext_vector_type(8))) float v8f;  // C/D frag 16x16 f32 (8 VGPRs)

// Problem constants: x(256,4,4,32,32) W(4,4,16,16,4,4,4,4,4) b(4,16,16,4)
// out(256,4,4,16,16).  Per (x,y): GEMM [256 x 1024] * [1024 x 16].
//
// Grid: 256 blocks (one per spatial pos). Block: 512 threads = 16 waves;
// wave w computes batch tile [16w, 16w+16) x all 16 (p,o) outputs.
// K = 1024 = 256 WMMA steps of K=4 (the 4 K-values of a step share (c,i,j,k)
// and sweep l=0..3, matching the ISA A-layout lane halves).

__global__ __launch_bounds__(512)
void ttn_pool_2x2_wmma(const float* __restrict__ x,
                       const float* __restrict__ W,
                       const float* __restrict__ bias,
                       float* __restrict__ out)
{
    // B fragments pre-swizzled: step s, lane L -> float2 at (s*32 + L).
    // v0/v1 hold rows Kbase+{0,1} (lanes 0-15) or Kbase+{2,3} (lanes 16-31).
    __shared__ float Blds[256 * 64];   // 64 KB

    const int tid = threadIdx.x;
    const int blk = blockIdx.x;        // 0..255
    const int px  = blk >> 4;          // x position 0..15
    const int py  = blk & 15;          // y position 0..15

    // -------- stage W slice (this x,y) into LDS in B-fragment order --------
    // W flat idx = c*1048576 + p*262144 + (px*16+py)*1024 + (K&255)*4 + o
    const long wxy = (long)(px * 16 + py) * 1024;
    #pragma unroll
    for (int it = 0; it < 8; ++it) {
        int idx = it * 512 + tid;              // 0..4095 : (p, K)
        int K   = idx & 1023;                  // contraction index (c,i,j,k,l)
        int p   = idx >> 10;                   // 0..3
        int c   = K >> 8;
        long g  = (long)c * 1048576 + (long)p * 262144 + wxy + (long)(K & 255) * 4;
        float4 w4 = *(const float4*)(W + g);   // o = 0..3, 16B aligned
        int s    = K >> 2;                     // WMMA step
        int r    = K & 3;                      // = l
        int base = s * 64 + ((r & 2) << 4) + (r & 1);   // half-wave + component
        Blds[base + (p * 4 + 0) * 2] = w4.x;   // N = p*4+o, float slot = base + N*2
        Blds[base + (p * 4 + 1) * 2] = w4.y;
        Blds[base + (p * 4 + 2) * 2] = w4.z;
        Blds[base + (p * 4 + 3) * 2] = w4.w;
    }
    __syncthreads();

    // -------- per-lane x quadrant values (A-fragment raw material) --------
    const int lane = tid & 31;
    const int wave = tid >> 5;          // batch tile 0..15
    const int hi   = lane >> 4;         // lane half: K offsets {0,1} vs {2,3}
    const int n    = wave * 16 + (lane & 15);   // A-matrix row M = lane&15

    // x flat idx = n*16384 + c*4096 + d*1024 + h*32 + w ; h0=2*px, w0=2*py
    const float* xb = x + (long)n * 16384 + (long)(px * 2) * 32 + (py * 2);
    float qa[16], qb[16], qc[16], qd[8];
    #pragma unroll
    for (int c = 0; c < 4; ++c) {
        #pragma unroll
        for (int i = 0; i < 4; ++i) {
            const float* xc = xb + c * 4096 + i * 1024;
            qa[c * 4 + i] = xc[0];      // a  = x[..., 2x,   2y  ]
            qb[c * 4 + i] = xc[32];     // b  = x[..., 2x+1, 2y  ]
            qc[c * 4 + i] = xc[1];      // c  = x[..., 2x,   2y+1]
        }
        // d = x[..., 2x+1, 2y+1]; this lane half only needs l = 2*hi, 2*hi+1
        qd[c * 2 + 0] = xb[c * 4096 + (2 * hi + 0) * 1024 + 33];
        qd[c * 2 + 1] = xb[c * 4096 + (2 * hi + 1) * 1024 + 33];
    }

    // -------- 256 x v_wmma_f32_16x16x4_f32 --------
    v8f acc = {0.f, 0.f, 0.f, 0.f, 0.f, 0.f, 0.f, 0.f};
    const float2* Bf = (const float2*)Blds;
    #pragma unroll
    for (int c = 0; c < 4; ++c) {
        float d0 = qd[c * 2 + 0];
        float d1 = qd[c * 2 + 1];
        #pragma unroll
        for (int i = 0; i < 4; ++i) {
            float t1 = qa[c * 4 + i];
            #pragma unroll
            for (int j = 0; j < 4; ++j) {
                float t2 = t1 * qb[c * 4 + j];
                #pragma unroll
                for (int k = 0; k < 4; ++k) {
                    float t3 = t2 * qc[c * 4 + k];
                    int s = ((c * 4 + i) * 4 + j) * 4 + k;
                    float2 bv = Bf[s * 32 + lane];          // ds_load_b64, conflict-free
                    v2f A = { t3 * d0, t3 * d1 };           // patch rows, K = 4s + l
                    v2f B = { bv.x, bv.y };
                    acc = __builtin_amdgcn_wmma_f32_16x16x4_f32(
                        /*neg_a=*/false, A, /*neg_b=*/false, B,
                        /*c_mod=*/(short)0, acc,
                        /*reuse_a=*/false, /*reuse_b=*/false);
                }
            }
        }
    }

    // -------- epilogue: bias + store C/D tile --------
    // C/D layout: lane L holds column N = L&15; VGPR v holds rows v (lanes<16)
    // or v+8 (lanes>=16). N = p*4 + o.
    const int N = lane & 15;
    const int p = N >> 2;
    const int o = N & 3;
    const float bv = bias[p * 1024 + px * 64 + py * 4 + o];
    // out flat idx = n*4096 + p*1024 + o*256 + px*16 + py
    const long obase = (long)p * 1024 + (long)o * 256 + px * 16 + py;
    #pragma unroll
    for (int v = 0; v < 8; ++v) {
        int on = wave * 16 + v + hi * 8;
        out[(long)on * 4096 + obase] = acc[v] + bv;
    }
}

extern "C" void kernel_launch(void* const* d_in, const int* in_sizes, int n_in,
                              void* d_out, int out_size, void* d_ws, size_t ws_size,
                              hipStream_t stream) {
    const float* x = (const float*)d_in[0];
    const float* W = (const float*)d_in[1];
    const float* b = (const float*)d_in[2];
    float* out     = (float*)d_out;
    ttn_pool_2x2_wmma<<<dim3(256), dim3(512), 0, stream>>>(x, W, b, out);
}